// EMLChain_25022479467179
// MI455X (gfx1250) — compile-verified
//
#include <hip/hip_runtime.h>
#include <math.h>

#define N_LEVELS 64
#define CLAMP_V  50.0
#define EPS2     1e-60               // EPS^2, EPS = 1e-30
#define LOG_EPS  -69.07755278982137  // log(1e-30)

#define BLOCK 256
#define ELEMS 4
#define TILE  (BLOCK * ELEMS)

// ln_logits staging: 63*2*3 doubles = 3024 bytes = 189 x b128 transfers
#define LN_BYTES 3024
#define LN_B128  189

// native clang vector type (accepted by __builtin_nontemporal_store)
typedef double v2d __attribute__((ext_vector_type(2)));

__global__ __launch_bounds__(BLOCK) void eml_chain_kernel(
    const double* __restrict__ x,        // [N]
    const double* __restrict__ l1,       // [2,2]
    const double* __restrict__ ln,       // [63,2,3]
    double*       __restrict__ out,      // [N] complex128 -> 2N doubles interleaved
    int n)
{
    __shared__ double raw[LN_BYTES / 8]; // staged ln_logits
    __shared__ double w[N_LEVELS][6];    // unified softmax weights

    const int t = threadIdx.x;
    const long base = (long)blockIdx.x * TILE + t;
    // uniform (scalar) full-tile test: avoids per-lane exec diamonds on the hot path
    const bool full_tile = ((long)(blockIdx.x + 1) * TILE) <= (long)n;

    // ---- issue global x loads FIRST so they overlap the weight-prep phase ----
    double xv[ELEMS], fre[ELEMS], fim[ELEMS];
    if (full_tile) {
#pragma unroll
        for (int e = 0; e < ELEMS; ++e)
            xv[e] = __builtin_nontemporal_load(&x[base + (long)e * BLOCK]);
    } else {
#pragma unroll
        for (int e = 0; e < ELEMS; ++e) {
            long idx = base + (long)e * BLOCK;
            xv[e] = (idx < n) ? x[idx] : 0.0;
        }
    }
#pragma unroll
    for (int e = 0; e < ELEMS; ++e) { fre[e] = 0.0; fim[e] = 0.0; }

    // ---- async-stage ln_logits into LDS (CDNA5 GLOBAL_LOAD_ASYNC_TO_LDS) ----
    {
        unsigned lds_base = (unsigned)(size_t)&raw[0];
        if (t < LN_B128) {
            unsigned voff    = (unsigned)t * 16u;  // per-lane global byte offset
            unsigned lds_dst = lds_base + voff;    // per-lane LDS byte address
            asm volatile("global_load_async_to_lds_b128 %0, %1, %2 offset:0"
                         :: "v"(lds_dst), "v"(voff), "s"(ln)
                         : "memory");
        }
        asm volatile("s_wait_asynccnt 0" ::: "memory");
    }
    __syncthreads();

    // ---- one thread per level computes the softmax weights into w[64][6] ----
    if (t < N_LEVELS) {
        if (t == 0) {
            // level 1: two softmax-of-2 rows; third weight forced to 0
            double a0 = l1[0], a1 = l1[1], b0 = l1[2], b1 = l1[3];
            double m  = fmax(a0, a1);
            double e0 = exp(a0 - m), e1 = exp(a1 - m);
            double s  = e0 + e1;
            w[0][0] = e0 / s; w[0][1] = e1 / s; w[0][2] = 0.0;
            m  = fmax(b0, b1);
            e0 = exp(b0 - m); e1 = exp(b1 - m);
            s  = e0 + e1;
            w[0][3] = e0 / s; w[0][4] = e1 / s; w[0][5] = 0.0;
        } else {
            const double* p = &raw[(t - 1) * 6];
#pragma unroll
            for (int r = 0; r < 2; ++r) {
                double a = p[3 * r + 0], b = p[3 * r + 1], c = p[3 * r + 2];
                double m  = fmax(fmax(a, b), c);
                double e0 = exp(a - m), e1 = exp(b - m), e2 = exp(c - m);
                double s  = e0 + e1 + e2;
                w[t][3 * r + 0] = e0 / s;
                w[t][3 * r + 1] = e1 / s;
                w[t][3 * r + 2] = e2 / s;
            }
        }
    }
    __syncthreads();

    // ---- main 64-level chain; 4 independent streams per thread for ILP ----
    for (int k = 0; k < N_LEVELS; ++k) {
        double w0 = w[k][0], w1 = w[k][1], w2 = w[k][2];
        double w3 = w[k][3], w4 = w[k][4], w5 = w[k][5];
#pragma unroll
        for (int e = 0; e < ELEMS; ++e) {
            // il = w0 + w1*x + w2*f ; ir = w3 + w4*x + w5*f  (x real)
            double ilr = fma(w2, fre[e], fma(w1, xv[e], w0));
            double ili = w2 * fim[e];
            double irr = fma(w5, fre[e], fma(w4, xv[e], w3));
            double iri = w5 * fim[e];

            // safe_exp(il): clamp real part, exp * (cos,sin)
            double ex = exp(fmin(fmax(ilr, -CLAMP_V), CLAMP_V));
            double sn, cs;
            sincos(ili, &sn, &cs);
            double exr = ex * cs;
            double exi = ex * sn;

            // safe_log(ir): |z| < EPS -> log(EPS) (constant); else complex log
            double m2   = fma(irr, irr, iri * iri);
            bool   tiny = (m2 < EPS2);
            double lre  = tiny ? LOG_EPS : 0.5 * log(m2);
            double lim  = tiny ? 0.0     : atan2(iri, irr);

            fre[e] = exr - lre;
            fim[e] = exi - lim;
        }
    }

    // ---- coalesced 16B (global_store_b128, non-temporal) interleaved stores ----
    v2d* o2 = (v2d*)out;
    if (full_tile) {
#pragma unroll
        for (int e = 0; e < ELEMS; ++e) {
            v2d v;
            v.x = fre[e];
            v.y = fim[e];
            __builtin_nontemporal_store(v, &o2[base + (long)e * BLOCK]);
        }
    } else {
#pragma unroll
        for (int e = 0; e < ELEMS; ++e) {
            long idx = base + (long)e * BLOCK;
            if (idx < n) {
                v2d v;
                v.x = fre[e];
                v.y = fim[e];
                o2[idx] = v;
            }
        }
    }
}

extern "C" void kernel_launch(void* const* d_in, const int* in_sizes, int n_in,
                              void* d_out, int out_size, void* d_ws, size_t ws_size,
                              hipStream_t stream) {
    const double* x  = (const double*)d_in[0];
    const double* l1 = (const double*)d_in[1];
    const double* ln = (const double*)d_in[2];
    double* out = (double*)d_out;

    int n = in_sizes[0];
    int blocks = (n + TILE - 1) / TILE;
    eml_chain_kernel<<<blocks, BLOCK, 0, stream>>>(x, l1, ln, out, n);
}